// GNNEncoder_52003464020252
// MI455X (gfx1250) — compile-verified
//
#include <hip/hip_runtime.h>
#include <math.h>

typedef __attribute__((ext_vector_type(2))) float v2f;
typedef __attribute__((ext_vector_type(8))) float v8f;
typedef unsigned int u32x4 __attribute__((ext_vector_type(4)));
typedef int i32x4 __attribute__((ext_vector_type(4)));
typedef int i32x8 __attribute__((ext_vector_type(8)));

#define IN_DIM  1024
#define N_ROW   8192
#define EMB     1024
#define KACT    8
#define NLAYER  3
#define N_NODES (IN_DIM + N_ROW)   // 9216
#define CSR_CAP 256

// ---------------------------------------------------------------------------
// h init: rows [0,IN_DIM) = emb_table, rows [IN_DIM, N_NODES) = 0
// ---------------------------------------------------------------------------
__global__ __launch_bounds__(256) void init_h(const float* __restrict__ emb,
                                              float* __restrict__ h) {
    size_t i = (size_t)blockIdx.x * 256 + threadIdx.x;
    h[i] = (i < (size_t)IN_DIM * EMB) ? emb[i] : 0.0f;
}

// ---------------------------------------------------------------------------
// Per-sample active-feature extraction (row-major nonzero order => ascending)
// ---------------------------------------------------------------------------
__global__ __launch_bounds__(256) void extract_idx(const int* __restrict__ x,
                                                   int* __restrict__ idx) {
    int r = blockIdx.x * 256 + threadIdx.x;
    if (r >= N_ROW) return;
    const int* xr = x + (size_t)r * IN_DIM;
    int j = 0;
    for (int c = 0; c < IN_DIM; ++c) {
        if (xr[c] != 0 && j < KACT) idx[r * KACT + j++] = c;
    }
    for (; j < KACT; ++j) idx[r * KACT + j] = 0;
}

// ---------------------------------------------------------------------------
// Transpose CSR (deterministic, ascending row order) + symmetric-norm dinv
// ---------------------------------------------------------------------------
__global__ __launch_bounds__(256) void build_csr(const int* __restrict__ x,
                                                 int* __restrict__ csr,
                                                 int* __restrict__ cnt,
                                                 float* __restrict__ dinv) {
    int c = blockIdx.x * 256 + threadIdx.x;
    if (c >= IN_DIM) return;
    int j = 0;
    for (int r = 0; r < N_ROW; ++r) {
        if (x[(size_t)r * IN_DIM + c] != 0 && j < CSR_CAP) csr[c * CSR_CAP + j++] = r;
    }
    cnt[c]  = j;
    dinv[c] = rsqrtf((float)(j + 1));
}

// ---------------------------------------------------------------------------
// TDM helper: DMA a 2D tile (tile1 rows x tile0 elems, f32) from global into
// LDS at byte offset lds_off, inserting pad_amount=4 dwords of LDS padding
// after every 2^(padc+1) dwords (padc=4 -> every 32 dwords, padc=5 -> 64).
// ---------------------------------------------------------------------------
__device__ __forceinline__ void tdm_load_2d(const void* gaddr, unsigned lds_off,
                                            unsigned tdim0, unsigned tdim1,
                                            unsigned tile0, unsigned tile1,
                                            unsigned stride0, unsigned padc) {
    unsigned long long ga = (unsigned long long)gaddr;
    u32x4 g0;
    g0.x = 1u;                                               // count=1, user mode
    g0.y = lds_off;                                          // lds_addr (bytes)
    g0.z = (unsigned)ga;                                     // global_addr[31:0]
    g0.w = (unsigned)((ga >> 32) & 0x1FFFFFFull) | (2u << 30); // addr[56:32] | type=2
    i32x8 g1;
    g1[0] = (int)((2u << 16)            // data_size = 4B
                | (1u << 20)            // pad_enable
                | (padc << 22)          // pad_interval
                | (3u << 25));          // pad_amount = 4 dwords
    g1[1] = (int)((tdim0 & 0xFFFFu) << 16);                        // tensor_dim0 lo
    g1[2] = (int)(((tdim0 >> 16) & 0xFFFFu) | ((tdim1 & 0xFFFFu) << 16));
    g1[3] = (int)(((tdim1 >> 16) & 0xFFFFu) | (tile0 << 16));      // tile_dim0
    g1[4] = (int)(tile1 & 0xFFFFu);                                // tile_dim1, tile_dim2=0
    g1[5] = (int)stride0;                                          // tensor_dim0_stride lo
    g1[6] = 0;
    g1[7] = 0;
    i32x4 gz = {0, 0, 0, 0};
#if defined(__clang_major__) && (__clang_major__ >= 23)
    i32x8 gz8 = {0, 0, 0, 0, 0, 0, 0, 0};
    __builtin_amdgcn_tensor_load_to_lds(g0, g1, gz, gz, gz8, 0);
#else
    __builtin_amdgcn_tensor_load_to_lds(g0, g1, gz, gz, 0);
#endif
}

// ---------------------------------------------------------------------------
// f32 WMMA GEMM, TDM-staged, double-buffered LDS.
// C[M,N] = A[M,K] * B[K,N], row-major; M%128==0, N%64==0, K%32==0.
// Block: 256 threads = 8 wave32; block tile 128x64; wave tile 32x32
// (2x2 of 16x16 WMMA tiles -> 4 V_WMMA_F32_16X16X4_F32 per k-step).
// ---------------------------------------------------------------------------
#define TM 128
#define TN 64
#define TK 32
#define A_FLOATS (TM * (TK + 4))        // 128 x 36 = 4608
#define B_FLOATS (TK * (TN + 4))        // 32 x 68  = 2176
#define STAGE_FLOATS (A_FLOATS + B_FLOATS)  // 6784

__global__ __launch_bounds__(256) void wmma_gemm_f32(const float* __restrict__ A,
                                                     const float* __restrict__ B,
                                                     float* __restrict__ C,
                                                     int M, int N, int K) {
    __shared__ float smem[2 * STAGE_FLOATS];   // single LDS block @ offset 0

    // --- compiler barrier for TDM-written LDS -------------------------------
    // The TDM engine writes smem, but the LDS address travels inside the D#
    // descriptor (an integer), so LLVM sees no stores to smem and would fold
    // all fragment loads to undef (observed round 2: ds=0, v_wmma on v[0:1]).
    // Inject an opaque, never-taken, in-bounds maybe-store so loads survive.
    int zoff;
    asm volatile("s_mov_b32 %0, 0" : "=s"(zoff));
    if (zoff & 1) smem[(zoff >> 1) & 4095] = 0.0f;
    // -----------------------------------------------------------------------

    const int tid  = threadIdx.x;
    const int lane = tid & 31;
    const int half = lane >> 4;        // 0: lanes 0-15, 1: lanes 16-31
    const int l16  = lane & 15;
    const int wv   = __builtin_amdgcn_readfirstlane(tid >> 5);  // uniform wave id

    const int bm = blockIdx.y * TM;
    const int bn = blockIdx.x * TN;

    const int wr = wv & 3;             // 32-row band: rows wr*32 .. +31
    const int wc = wv >> 2;            // 32-col band: cols wc*32 .. +31

    v8f acc00 = {}, acc01 = {}, acc10 = {}, acc11 = {};

    const int NCHUNK = K / TK;

    // preload chunk 0 into stage 0 (wave 0 only; uniform scalar branch)
    if (wv == 0) {
        tdm_load_2d(A + (size_t)bm * K, 0u,
                    (unsigned)K, (unsigned)M, TK, TM, (unsigned)K, 4u);
        tdm_load_2d(B + bn, (unsigned)(A_FLOATS * 4),
                    (unsigned)N, (unsigned)K, TN, TK, (unsigned)N, 5u);
    }

    const int arow0 = (wr * 32 + l16) * (TK + 4);
    const int arow1 = arow0 + 16 * (TK + 4);
    const int cb0   = wc * 32 + l16;
    const int cb1   = cb0 + 16;

    for (int ch = 0; ch < NCHUNK; ++ch) {
        const int s = ch & 1;
        if (wv == 0) {
            if (ch + 1 < NCHUNK) {
                const int k1 = (ch + 1) * TK;
                const unsigned sb = (unsigned)((s ^ 1) * STAGE_FLOATS * 4);
                tdm_load_2d(A + (size_t)bm * K + k1, sb,
                            (unsigned)K, (unsigned)M, TK, TM, (unsigned)K, 4u);
                tdm_load_2d(B + (size_t)k1 * N + bn, sb + (unsigned)(A_FLOATS * 4),
                            (unsigned)N, (unsigned)K, TN, TK, (unsigned)N, 5u);
                __builtin_amdgcn_s_wait_tensorcnt(2);   // current stage's pair done
            } else {
                __builtin_amdgcn_s_wait_tensorcnt(0);
            }
        }
        __syncthreads();                                // stage s visible to all waves

        const float* as = smem + s * STAGE_FLOATS;      // [128][36]
        const float* bs = as + A_FLOATS;                // [32][68]

#pragma unroll
        for (int kk = 0; kk < TK; kk += 4) {
            const int ka = kk + half * 2;   // lanes 0-15: K={kk,kk+1}; 16-31: {kk+2,kk+3}
            v2f a0, a1, b0, b1;
            a0.x = as[arow0 + ka];           a0.y = as[arow0 + ka + 1];
            a1.x = as[arow1 + ka];           a1.y = as[arow1 + ka + 1];
            b0.x = bs[ka * (TN + 4) + cb0];  b0.y = bs[(ka + 1) * (TN + 4) + cb0];
            b1.x = bs[ka * (TN + 4) + cb1];  b1.y = bs[(ka + 1) * (TN + 4) + cb1];

            acc00 = __builtin_amdgcn_wmma_f32_16x16x4_f32(false, a0, false, b0,
                                                          (short)0, acc00, false, false);
            acc01 = __builtin_amdgcn_wmma_f32_16x16x4_f32(false, a0, false, b1,
                                                          (short)0, acc01, false, false);
            acc10 = __builtin_amdgcn_wmma_f32_16x16x4_f32(false, a1, false, b0,
                                                          (short)0, acc10, false, false);
            acc11 = __builtin_amdgcn_wmma_f32_16x16x4_f32(false, a1, false, b1,
                                                          (short)0, acc11, false, false);
        }
        __syncthreads();                    // all waves done reading stage s
    }

    // D layout: VGPR i -> row i (lanes 0-15) / i+8 (lanes 16-31); col = lane%16
#pragma unroll
    for (int i = 0; i < 8; ++i) {
        const int r0 = bm + wr * 32 + i + half * 8;
        const int r1 = r0 + 16;
        const int c0 = bn + wc * 32 + l16;
        C[(size_t)r0 * N + c0]      = acc00[i];
        C[(size_t)r0 * N + c0 + 16] = acc01[i];
        C[(size_t)r1 * N + c0]      = acc10[i];
        C[(size_t)r1 * N + c0 + 16] = acc11[i];
    }
}

// ---------------------------------------------------------------------------
// Feature-node aggregation: h_new[c] = dinv[c]^2*hW[c] + dinv[c]*ds*sum hW[r] + b
// ---------------------------------------------------------------------------
__global__ __launch_bounds__(256) void feat_agg(const float* __restrict__ hW,
                                                const int* __restrict__ csr,
                                                const int* __restrict__ cnt,
                                                const float* __restrict__ dinv,
                                                const float* __restrict__ bias,
                                                float* __restrict__ hout) {
    const int c = blockIdx.y;
    const int d = blockIdx.x * 256 + threadIdx.x;
    const float ds = rsqrtf(9.0f);
    const float dc = dinv[c];
    const int n = cnt[c];
    float s = 0.0f;
    for (int j = 0; j < n; ++j) {
        const int r = csr[c * CSR_CAP + j];
        s += hW[(size_t)(IN_DIM + r) * EMB + d];
    }
    hout[(size_t)c * EMB + d] = dc * dc * hW[(size_t)c * EMB + d] + dc * ds * s + bias[d];
}

// ---------------------------------------------------------------------------
// Sample-node aggregation: h_new[r] = ds^2*hW[r] + ds*sum_k dinv[c_k]*hW[c_k] + b
// ---------------------------------------------------------------------------
__global__ __launch_bounds__(256) void samp_agg(const float* __restrict__ hW,
                                                const int* __restrict__ idx,
                                                const float* __restrict__ dinv,
                                                const float* __restrict__ bias,
                                                float* __restrict__ out) {
    const int r = blockIdx.y;
    const int d = blockIdx.x * 256 + threadIdx.x;
    const float ds = rsqrtf(9.0f);
    float acc = ds * ds * hW[(size_t)(IN_DIM + r) * EMB + d];
#pragma unroll
    for (int k = 0; k < KACT; ++k) {
        const int c = idx[r * KACT + k];
        acc += ds * dinv[c] * hW[(size_t)c * EMB + d];
    }
    out[(size_t)r * EMB + d] = acc + bias[d];
}

// ---------------------------------------------------------------------------
// Launch
// ---------------------------------------------------------------------------
extern "C" void kernel_launch(void* const* d_in, const int* in_sizes, int n_in,
                              void* d_out, int out_size, void* d_ws, size_t ws_size,
                              hipStream_t stream) {
    const int*   x   = (const int*)d_in[0];
    const float* emb = (const float*)d_in[1];
    const float* Ws  = (const float*)d_in[2];
    const float* bs  = (const float*)d_in[3];
    (void)in_sizes; (void)n_in; (void)out_size; (void)ws_size;

    const size_t Helems = (size_t)N_NODES * EMB;
    float* buf0 = (float*)d_ws;                          // h (layer input / agg output)
    float* buf1 = buf0 + Helems;                         // hW (GEMM output)
    int*   idx  = (int*)(buf1 + Helems);                 // [N_ROW * KACT]
    int*   csr  = idx + (size_t)N_ROW * KACT;            // [IN_DIM * CSR_CAP]
    int*   cnt  = csr + (size_t)IN_DIM * CSR_CAP;        // [IN_DIM]
    float* dinv = (float*)(cnt + IN_DIM);                // [IN_DIM]

    init_h<<<(unsigned)(Helems / 256), 256, 0, stream>>>(emb, buf0);
    extract_idx<<<(N_ROW + 255) / 256, 256, 0, stream>>>(x, idx);
    build_csr<<<(IN_DIM + 255) / 256, 256, 0, stream>>>(x, csr, cnt, dinv);

    for (int li = 0; li < NLAYER; ++li) {
        wmma_gemm_f32<<<dim3(EMB / TN, N_NODES / TM), 256, 0, stream>>>(
            buf0, Ws + (size_t)li * EMB * EMB, buf1, N_NODES, EMB, EMB);
        const float* b = bs + (size_t)li * EMB;
        if (li < NLAYER - 1) {
            feat_agg<<<dim3(EMB / 256, IN_DIM), 256, 0, stream>>>(buf1, csr, cnt, dinv, b, buf0);
            samp_agg<<<dim3(EMB / 256, N_ROW), 256, 0, stream>>>(
                buf1, idx, dinv, b, buf0 + (size_t)IN_DIM * EMB);
        } else {
            samp_agg<<<dim3(EMB / 256, N_ROW), 256, 0, stream>>>(
                buf1, idx, dinv, b, (float*)d_out);
        }
    }
}